// EAMForce_11854109737005
// MI455X (gfx1250) — compile-verified
//
#include <hip/hip_runtime.h>
#include <hip/hip_bf16.h>

// ---------------------------------------------------------------------------
// EAM forces on MI455X (gfx1250): gather/scatter bound, everything gathered
// (positions/types/tables/rho/forces) is L2-resident (192MB L2). Strategy:
//  - pack (x,y,z,type) into one aligned float4 -> single b128 gather/endpoint
//  - native f32 global atomics (relaxed, agent scope -> global_atomic_add_f32)
//  - 2 independent pairs per thread for memory-level parallelism
//  - WGP-scope global_prefetch_b8 on the streamed edge lists
//  - RT caching so the force pass re-reads edges from L2 (51MB << 192MB)
// ---------------------------------------------------------------------------

#define ETYPES 2
#define NR     8192
#define NRHO   4096
#define RMAXF  6.0f
#define EPSF   1e-7f
#define INVDR  ((float)(NR - 1) / RMAXF)
#define TB     256        // 8 wave32 waves per block
#define PFD    8192       // edge-list prefetch distance (pairs)

static __device__ __forceinline__ void atom_add_f32(float* p, float v) {
    __hip_atomic_fetch_add(p, v, __ATOMIC_RELAXED, __HIP_MEMORY_SCOPE_AGENT);
}

// ---- lowering probe: FIRST kernel in TU so the asm snippet shows whether
// ---- the f32 atomic is a native global_atomic_add_f32 or a CAS loop.
__global__ void eam_atomic_probe(float* p, float v) {
    atom_add_f32(p + threadIdx.x, v);
}

static __device__ __forceinline__ float interp1(const float* __restrict__ tab,
                                                int idx, float fr) {
    float t0 = tab[idx];
    float t1 = tab[idx + 1];          // idx <= row_end-1 guaranteed by clip
    return fmaf(fr, t1 - t0, t0);
}

// ---------------------- packed pair kernels (hot path) ---------------------

static __device__ __forceinline__ void density_one(
    const float4* __restrict__ pos4, const float* __restrict__ dens_tab,
    const int* __restrict__ ei, const int* __restrict__ ej,
    float* __restrict__ rho, int p) {
    int i = ei[p];
    int j = ej[p];
    float4 Pi = pos4[i];              // global_load_b128 (pos + type)
    float4 Pj = pos4[j];
    float dx = Pj.x - Pi.x, dy = Pj.y - Pi.y, dz = Pj.z - Pi.z;
    float s2 = fmaf(dx, dx, fmaf(dy, dy, fmaf(dz, dz, 1e-12f)));
    float r  = sqrtf(s2);
    float rc = fminf(r, RMAXF * (1.0f - EPSF));
    float f  = rc * INVDR;
    int   i0 = (int)f;                // in [0, NR-2]
    float fr = f - (float)i0;
    int ti = __float_as_int(Pi.w);
    int tj = __float_as_int(Pj.w);
    float dens_j = interp1(dens_tab, tj * NR + i0, fr);
    float dens_i = interp1(dens_tab, ti * NR + i0, fr);
    atom_add_f32(rho + i, dens_j);
    atom_add_f32(rho + j, dens_i);
}

__global__ void eam_density_kernel(const float4* __restrict__ pos4,
                                   const float* __restrict__ dens_tab,
                                   const int*   __restrict__ ei,
                                   const int*   __restrict__ ej,
                                   float*       __restrict__ rho,
                                   int n_pairs) {
    int base = blockIdx.x * (2 * TB) + threadIdx.x;
    int p1   = base + TB;
    int pf   = base + PFD;
    if (pf + TB < n_pairs) {
        __builtin_prefetch(ei + pf, 0, 3);        // WGP-scope global_prefetch_b8
        __builtin_prefetch(ej + pf, 0, 3);
        __builtin_prefetch(ei + pf + TB, 0, 3);
        __builtin_prefetch(ej + pf + TB, 0, 3);
    }
    if (p1 < n_pairs) {               // hot: two independent gather chains
        density_one(pos4, dens_tab, ei, ej, rho, base);
        density_one(pos4, dens_tab, ei, ej, rho, p1);
    } else if (base < n_pairs) {
        density_one(pos4, dens_tab, ei, ej, rho, base);
    }
}

static __device__ __forceinline__ void force_one(
    const float4* __restrict__ pos4, const float* __restrict__ dd_tab,
    const float* __restrict__ dphi_tab, const float* __restrict__ dF,
    const int* __restrict__ ei, const int* __restrict__ ej,
    float* __restrict__ forces, int p) {
    int i = ei[p];
    int j = ej[p];
    float4 Pi = pos4[i];
    float4 Pj = pos4[j];
    float dx = Pj.x - Pi.x, dy = Pj.y - Pi.y, dz = Pj.z - Pi.z;
    float s2 = fmaf(dx, dx, fmaf(dy, dy, fmaf(dz, dz, 1e-12f)));
    float r  = sqrtf(s2);
    float rc = fminf(r, RMAXF * (1.0f - EPSF));
    float f  = rc * INVDR;
    int   i0 = (int)f;
    float fr = f - (float)i0;
    int ti = __float_as_int(Pi.w);
    int tj = __float_as_int(Pj.w);
    float ddens_j = interp1(dd_tab, tj * NR + i0, fr);
    float ddens_i = interp1(dd_tab, ti * NR + i0, fr);
    float dphi    = interp1(dphi_tab, (ti * ETYPES + tj) * NR + i0, fr);
    float coeff = fmaf(dF[i], ddens_j, fmaf(dF[j], ddens_i, dphi));
    float c  = coeff / r;
    float fx = c * dx, fy = c * dy, fz = c * dz;
    atom_add_f32(forces + 3 * i + 0, -fx);
    atom_add_f32(forces + 3 * i + 1, -fy);
    atom_add_f32(forces + 3 * i + 2, -fz);
    atom_add_f32(forces + 3 * j + 0,  fx);
    atom_add_f32(forces + 3 * j + 1,  fy);
    atom_add_f32(forces + 3 * j + 2,  fz);
}

__global__ void eam_force_kernel(const float4* __restrict__ pos4,
                                 const float* __restrict__ dd_tab,
                                 const float* __restrict__ dphi_tab,
                                 const float* __restrict__ dF,
                                 const int*   __restrict__ ei,
                                 const int*   __restrict__ ej,
                                 float*       __restrict__ forces,
                                 int n_pairs) {
    int base = blockIdx.x * (2 * TB) + threadIdx.x;
    int p1   = base + TB;
    int pf   = base + PFD;
    if (pf + TB < n_pairs) {
        __builtin_prefetch(ei + pf, 0, 3);
        __builtin_prefetch(ej + pf, 0, 3);
        __builtin_prefetch(ei + pf + TB, 0, 3);
        __builtin_prefetch(ej + pf + TB, 0, 3);
    }
    if (p1 < n_pairs) {
        force_one(pos4, dd_tab, dphi_tab, dF, ei, ej, forces, base);
        force_one(pos4, dd_tab, dphi_tab, dF, ei, ej, forces, p1);
    } else if (base < n_pairs) {
        force_one(pos4, dd_tab, dphi_tab, dF, ei, ej, forces, base);
    }
}

// --------------------------- prep / embed kernels --------------------------

__global__ void eam_pack_zero_kernel(const float* __restrict__ pos,
                                     const int*   __restrict__ types,
                                     float4*      __restrict__ pos4,
                                     float*       __restrict__ rho,
                                     float*       __restrict__ forces,
                                     int n_atoms) {
    int i = blockIdx.x * blockDim.x + threadIdx.x;
    if (i < n_atoms) {
        float4 v;
        v.x = pos[3 * i + 0];
        v.y = pos[3 * i + 1];
        v.z = pos[3 * i + 2];
        v.w = __int_as_float(types[i]);
        pos4[i] = v;
        rho[i]  = 0.0f;
    }
    if (i < 3 * n_atoms) forces[i] = 0.0f;
}

__global__ void eam_embed_kernel(const float* __restrict__ embed_tab,
                                 const float* __restrict__ rho_min,
                                 const float* __restrict__ inv_drho,
                                 const int*   __restrict__ types,
                                 const float* __restrict__ rho,
                                 float*       __restrict__ dF,
                                 int n_atoms) {
    int n = blockIdx.x * blockDim.x + threadIdx.x;
    if (n >= n_atoms) return;
    int   t    = types[n];
    float rmin = rho_min[t];
    float invd = inv_drho[t];
    float hi   = rmin + (float)(NRHO - 1) / invd * (1.0f - EPSF);
    float rc   = fminf(fmaxf(rho[n], rmin), hi);
    float g    = (rc - rmin) * invd;
    int   g0   = (int)g;
    g0 = max(0, min(g0, NRHO - 2));
    float gfr  = g - (float)g0;
    dF[n] = interp1(embed_tab, t * NRHO + g0, gfr);
}

// --------------------- unpacked fallback (small d_ws) ----------------------

__global__ void eam_zero_kernel(float* __restrict__ rho,
                                float* __restrict__ forces, int n_atoms) {
    int i = blockIdx.x * blockDim.x + threadIdx.x;
    if (i < n_atoms)     rho[i]    = 0.0f;
    if (i < 3 * n_atoms) forces[i] = 0.0f;
}

__global__ void eam_density_kernel_u(const float* __restrict__ pos,
                                     const float* __restrict__ dens_tab,
                                     const int*   __restrict__ types,
                                     const int*   __restrict__ ei,
                                     const int*   __restrict__ ej,
                                     float*       __restrict__ rho,
                                     int n_pairs) {
    int p = blockIdx.x * blockDim.x + threadIdx.x;
    if (p >= n_pairs) return;
    int i = ei[p], j = ej[p];
    float xi = pos[3 * i], yi = pos[3 * i + 1], zi = pos[3 * i + 2];
    float xj = pos[3 * j], yj = pos[3 * j + 1], zj = pos[3 * j + 2];
    float dx = xj - xi, dy = yj - yi, dz = zj - zi;
    float r  = sqrtf(fmaf(dx, dx, fmaf(dy, dy, fmaf(dz, dz, 1e-12f))));
    float f  = fminf(r, RMAXF * (1.0f - EPSF)) * INVDR;
    int   i0 = (int)f;
    float fr = f - (float)i0;
    int ti = types[i], tj = types[j];
    atom_add_f32(rho + i, interp1(dens_tab, tj * NR + i0, fr));
    atom_add_f32(rho + j, interp1(dens_tab, ti * NR + i0, fr));
}

__global__ void eam_force_kernel_u(const float* __restrict__ pos,
                                   const float* __restrict__ dd_tab,
                                   const float* __restrict__ dphi_tab,
                                   const int*   __restrict__ types,
                                   const int*   __restrict__ ei,
                                   const int*   __restrict__ ej,
                                   const float* __restrict__ dF,
                                   float*       __restrict__ forces,
                                   int n_pairs) {
    int p = blockIdx.x * blockDim.x + threadIdx.x;
    if (p >= n_pairs) return;
    int i = ei[p], j = ej[p];
    float xi = pos[3 * i], yi = pos[3 * i + 1], zi = pos[3 * i + 2];
    float xj = pos[3 * j], yj = pos[3 * j + 1], zj = pos[3 * j + 2];
    float dx = xj - xi, dy = yj - yi, dz = zj - zi;
    float r  = sqrtf(fmaf(dx, dx, fmaf(dy, dy, fmaf(dz, dz, 1e-12f))));
    float f  = fminf(r, RMAXF * (1.0f - EPSF)) * INVDR;
    int   i0 = (int)f;
    float fr = f - (float)i0;
    int ti = types[i], tj = types[j];
    float coeff = fmaf(dF[i], interp1(dd_tab, tj * NR + i0, fr),
                  fmaf(dF[j], interp1(dd_tab, ti * NR + i0, fr),
                       interp1(dphi_tab, (ti * ETYPES + tj) * NR + i0, fr)));
    float c = coeff / r;
    float fx = c * dx, fy = c * dy, fz = c * dz;
    atom_add_f32(forces + 3 * i + 0, -fx);
    atom_add_f32(forces + 3 * i + 1, -fy);
    atom_add_f32(forces + 3 * i + 2, -fz);
    atom_add_f32(forces + 3 * j + 0,  fx);
    atom_add_f32(forces + 3 * j + 1,  fy);
    atom_add_f32(forces + 3 * j + 2,  fz);
}

// ------------------------------- launcher ----------------------------------

extern "C" void kernel_launch(void* const* d_in, const int* in_sizes, int n_in,
                              void* d_out, int out_size, void* d_ws, size_t ws_size,
                              hipStream_t stream) {
    const float* positions     = (const float*)d_in[0];
    const float* density_tab   = (const float*)d_in[1];
    const float* density_deriv = (const float*)d_in[2];
    const float* pair_deriv    = (const float*)d_in[3];
    const float* embed_deriv   = (const float*)d_in[4];
    const float* embed_rho_min = (const float*)d_in[5];
    const float* embed_invdrho = (const float*)d_in[6];
    const int*   atom_types    = (const int*)d_in[7];
    const int*   edge_i        = (const int*)d_in[8];
    const int*   edge_j        = (const int*)d_in[9];

    const int n_atoms = in_sizes[7];
    const int n_pairs = in_sizes[8];
    float* forces = (float*)d_out;

    const size_t need_packed = (size_t)n_atoms * 6 * sizeof(float);  // pos4 + rho + dF

    if (ws_size >= need_packed) {
        // layout: pos4 [4N floats, 16B aligned] | rho [N] | dF [N]
        float4* pos4 = (float4*)d_ws;
        float*  rho  = (float*)(pos4 + n_atoms);
        float*  dF   = rho + n_atoms;

        eam_pack_zero_kernel<<<(3 * n_atoms + TB - 1) / TB, TB, 0, stream>>>(
            positions, atom_types, pos4, rho, forces, n_atoms);

        eam_density_kernel<<<(n_pairs + 2 * TB - 1) / (2 * TB), TB, 0, stream>>>(
            pos4, density_tab, edge_i, edge_j, rho, n_pairs);

        eam_embed_kernel<<<(n_atoms + TB - 1) / TB, TB, 0, stream>>>(
            embed_deriv, embed_rho_min, embed_invdrho, atom_types, rho, dF, n_atoms);

        eam_force_kernel<<<(n_pairs + 2 * TB - 1) / (2 * TB), TB, 0, stream>>>(
            pos4, density_deriv, pair_deriv, dF, edge_i, edge_j, forces, n_pairs);
    } else {
        // fallback: rho [N] | dF [N]
        float* rho = (float*)d_ws;
        float* dF  = rho + n_atoms;

        eam_zero_kernel<<<(3 * n_atoms + TB - 1) / TB, TB, 0, stream>>>(
            rho, forces, n_atoms);

        eam_density_kernel_u<<<(n_pairs + TB - 1) / TB, TB, 0, stream>>>(
            positions, density_tab, atom_types, edge_i, edge_j, rho, n_pairs);

        eam_embed_kernel<<<(n_atoms + TB - 1) / TB, TB, 0, stream>>>(
            embed_deriv, embed_rho_min, embed_invdrho, atom_types, rho, dF, n_atoms);

        eam_force_kernel_u<<<(n_pairs + TB - 1) / TB, TB, 0, stream>>>(
            positions, density_deriv, pair_deriv, atom_types, edge_i, edge_j,
            dF, forces, n_pairs);
    }
}